// RotatedIoULoss_18485539242565
// MI455X (gfx1250) — compile-verified
//
#include <hip/hip_runtime.h>
#include <math.h>

#define EPSF      1e-8f
#define IN_TOLF   1e-6f
#define LOSS_EPSF 1e-6f
#define NCAND 24
#define BLOCK 128   // 4 waves; multiple of 64 -> LDS column layout is bank-conflict-free

__global__ void riou_init_kernel(double* __restrict__ acc) {
    if (threadIdx.x == 0 && blockIdx.x == 0) acc[0] = 0.0;
}

__global__ void riou_final_kernel(const double* __restrict__ acc,
                                  float* __restrict__ out, int N) {
    if (threadIdx.x == 0 && blockIdx.x == 0)
        out[0] = (float)(acc[0] / (double)N);
}

__global__ __launch_bounds__(BLOCK)
void riou_main_kernel(const float* __restrict__ pred,
                      const float* __restrict__ target,
                      double* __restrict__ acc, int N)
{
    // Per-thread candidate arrays live in LDS (not scratch):
    // layout [k][tid] -> bank = (k*BLOCK + tid) % 64 = tid % 64, so each lane
    // owns a fixed bank independent of the (divergent) index k.
    __shared__ float s_vx [NCAND * BLOCK];
    __shared__ float s_vy [NCAND * BLOCK];
    __shared__ float s_ang[NCAND * BLOCK];
    __shared__ float wsum[BLOCK / 32];

    const int tid = threadIdx.x;
    const int gid = blockIdx.x * BLOCK + tid;

#define VX(k)  s_vx [(k) * BLOCK + tid]
#define VY(k)  s_vy [(k) * BLOCK + tid]
#define ANG(k) s_ang[(k) * BLOCK + tid]

    float loss = 0.0f;

    if (gid < N) {
        const float* p = pred   + (size_t)gid * 5u;
        const float* t = target + (size_t)gid * 5u;
        __builtin_prefetch(t, 0, 3);   // gfx1250: global_prefetch_b8

        const float px = p[0], py = p[1], pw = p[2], ph = p[3], pa = p[4];
        const float tx = t[0], ty = t[1], tw = t[2], th = t[3], ta = t[4];

        // ---- corners (reference dx/dy ordering) ----
        const float DX[4] = { 0.5f, -0.5f, -0.5f,  0.5f };
        const float DY[4] = { 0.5f,  0.5f, -0.5f, -0.5f };
        float s1, c1, s2, c2;
        sincosf(pa, &s1, &c1);
        sincosf(ta, &s2, &c2);

        float ax4[4], ay4[4], bx4[4], by4[4];
        #pragma unroll
        for (int k = 0; k < 4; ++k) {
            const float dx1 = DX[k] * pw, dy1 = DY[k] * ph;
            ax4[k] = dx1 * c1 - dy1 * s1 + px;
            ay4[k] = dx1 * s1 + dy1 * c1 + py;
            const float dx2 = DX[k] * tw, dy2 = DY[k] * th;
            bx4[k] = dx2 * c2 - dy2 * s2 + tx;
            by4[k] = dx2 * s2 + dy2 * c2 + ty;
        }

        // ---- compact active vertices (order: c1 corners, c2 corners, ipts) ----
        int   cnt = 0;
        float sx = 0.0f, sy = 0.0f;

        // corners of box1 inside box2
        {
            const float abx = bx4[1] - bx4[0], aby = by4[1] - by4[0];
            const float adx = bx4[3] - bx4[0], ady = by4[3] - by4[0];
            const float iab = 1.0f / (abx * abx + aby * aby);
            const float iad = 1.0f / (adx * adx + ady * ady);
            #pragma unroll
            for (int k = 0; k < 4; ++k) {
                const float amx = ax4[k] - bx4[0], amy = ay4[k] - by4[0];
                const float pab = (abx * amx + aby * amy) * iab;
                const float pad = (adx * amx + ady * amy) * iad;
                if (pab > -IN_TOLF && pab < 1.0f + IN_TOLF &&
                    pad > -IN_TOLF && pad < 1.0f + IN_TOLF) {
                    VX(cnt) = ax4[k]; VY(cnt) = ay4[k];
                    sx += ax4[k]; sy += ay4[k]; ++cnt;
                }
            }
        }
        // corners of box2 inside box1
        {
            const float abx = ax4[1] - ax4[0], aby = ay4[1] - ay4[0];
            const float adx = ax4[3] - ax4[0], ady = ay4[3] - ay4[0];
            const float iab = 1.0f / (abx * abx + aby * aby);
            const float iad = 1.0f / (adx * adx + ady * ady);
            #pragma unroll
            for (int k = 0; k < 4; ++k) {
                const float amx = bx4[k] - ax4[0], amy = by4[k] - ay4[0];
                const float pab = (abx * amx + aby * amy) * iab;
                const float pad = (adx * amx + ady * amy) * iad;
                if (pab > -IN_TOLF && pab < 1.0f + IN_TOLF &&
                    pad > -IN_TOLF && pad < 1.0f + IN_TOLF) {
                    VX(cnt) = bx4[k]; VY(cnt) = by4[k];
                    sx += bx4[k]; sy += by4[k]; ++cnt;
                }
            }
        }
        // edge-edge intersections (i-major, j-minor to match reference flatten)
        #pragma unroll
        for (int i0 = 0; i0 < 4; ++i0) {
            const float x1 = ax4[i0],           y1 = ay4[i0];
            const float x2 = ax4[(i0 + 1) & 3], y2 = ay4[(i0 + 1) & 3];
            const float ex = x2 - x1, ey = y2 - y1;
            #pragma unroll
            for (int j0 = 0; j0 < 4; ++j0) {
                const float x3 = bx4[j0],           y3 = by4[j0];
                const float x4 = bx4[(j0 + 1) & 3], y4 = by4[(j0 + 1) & 3];
                const float fx = x4 - x3, fy = y4 - y3;
                const float num   = fy * ex - fx * ey;
                const float den_t = fx * (y1 - y3) - fy * (x1 - x3);
                const float den_u = ex * (y1 - y3) - ey * (x1 - x3);
                bool m;
                if (num == 0.0f) {
                    m = false;
                } else {
                    const float t_m = den_t / num;
                    const float u_m = den_u / num;
                    m = (t_m > 0.0f) && (t_m < 1.0f) && (u_m > 0.0f) && (u_m < 1.0f);
                }
                if (m) {
                    const float tt = den_t / (num + EPSF);
                    const float ix = x1 + tt * ex;
                    const float iy = y1 + tt * ey;
                    VX(cnt) = ix; VY(cnt) = iy;
                    sx += ix; sy += iy; ++cnt;
                }
            }
        }

        // ---- center at mean of active vertices, compute angles ----
        const float nv = (float)(cnt > 0 ? cnt : 1);
        const float mx = sx / nv, my = sy / nv;
        for (int k = 0; k < cnt; ++k) {
            const float cvx = VX(k) - mx;
            const float cvy = VY(k) - my;
            VX(k) = cvx; VY(k) = cvy;
            ANG(k) = atan2f(cvy, cvx);
        }

        // ---- stable insertion sort by angle (matches stable argsort) ----
        for (int k = 1; k < cnt; ++k) {
            const float ka = ANG(k), kx = VX(k), ky = VY(k);
            int j = k - 1;
            while (j >= 0 && ANG(j) > ka) {
                ANG(j + 1) = ANG(j); VX(j + 1) = VX(j); VY(j + 1) = VY(j);
                --j;
            }
            ANG(j + 1) = ka; VX(j + 1) = kx; VY(j + 1) = ky;
        }

        // ---- shoelace ----
        float cs = 0.0f;
        for (int k = 0; k < cnt; ++k) {
            const int kn = (k + 1 == cnt) ? 0 : k + 1;
            cs += VX(k) * VY(kn) - VY(k) * VX(kn);
        }
        const float inter = fabsf(cs) * 0.5f;

        const float area1 = pw * ph;
        const float area2 = tw * th;
        float iou = inter / (area1 + area2 - inter);
        iou = fminf(fmaxf(iou, 0.0f), 1.0f);
        iou = fmaxf(iou, LOSS_EPSF);
        loss = -logf(iou);
    }

#undef VX
#undef VY
#undef ANG

    // ---- wave32 reduction -> LDS -> block sum -> f64 global atomic ----
    #pragma unroll
    for (int off = 16; off > 0; off >>= 1)
        loss += __shfl_down(loss, off, 32);

    const int lane = tid & 31;
    const int wid  = tid >> 5;
    if (lane == 0) wsum[wid] = loss;
    __syncthreads();

    if (wid == 0) {
        float v = (lane < (BLOCK / 32)) ? wsum[lane] : 0.0f;
        #pragma unroll
        for (int off = (BLOCK / 64); off > 0; off >>= 1)
            v += __shfl_down(v, off, 32);
        if (lane == 0)
            atomicAdd(acc, (double)v);   // gfx1250: global_atomic_add_f64
    }
}

extern "C" void kernel_launch(void* const* d_in, const int* in_sizes, int n_in,
                              void* d_out, int out_size, void* d_ws, size_t ws_size,
                              hipStream_t stream) {
    const float* pred   = (const float*)d_in[0];
    const float* target = (const float*)d_in[1];
    float*  out = (float*)d_out;
    double* acc = (double*)d_ws;

    const int N = in_sizes[0] / 5;
    const int grid = (N + BLOCK - 1) / BLOCK;

    riou_init_kernel<<<1, 1, 0, stream>>>(acc);
    riou_main_kernel<<<grid, BLOCK, 0, stream>>>(pred, target, acc, N);
    riou_final_kernel<<<1, 1, 0, stream>>>(acc, out, N);
}